// EdgeAttentionLayer_14173392077054
// MI455X (gfx1250) — compile-verified
//
#include <hip/hip_runtime.h>

#define F_DIM 64
#define H_DIM 4

typedef __attribute__((ext_vector_type(2))) float v2f;
typedef __attribute__((ext_vector_type(8))) float v8f;

// ---------- native-atomic helpers ----------
__device__ __forceinline__ void atomic_add_f32(float* addr, float val) {
  // lowers to GLOBAL_ATOMIC_ADD_F32 (no return -> STOREcnt path)
  __hip_atomic_fetch_add(addr, val, __ATOMIC_RELAXED, __HIP_MEMORY_SCOPE_AGENT);
}

__device__ __forceinline__ void atomic_max_f32(float* addr, float val) {
  // IEEE-754 order-preserving integer punning:
  //   nonneg floats: monotone as signed int -> atomic smax
  //   negative floats: reverse-monotone as unsigned -> atomic umin
  if (val >= 0.0f) {
    __hip_atomic_fetch_max((int*)addr, __float_as_int(val),
                           __ATOMIC_RELAXED, __HIP_MEMORY_SCOPE_AGENT);
  } else {
    __hip_atomic_fetch_min((unsigned int*)addr, __float_as_uint(val),
                           __ATOMIC_RELAXED, __HIP_MEMORY_SCOPE_AGENT);
  }
}

// ---------- K0: zero output, init segment max/sum ----------
__global__ __launch_bounds__(256)
void init_kernel(float* __restrict__ out, float* __restrict__ m,
                 float* __restrict__ s, int NF, int NH) {
  int i = blockIdx.x * blockDim.x + threadIdx.x;
  if (i < NF) out[i] = 0.0f;
  if (i < NH) {
    m[i] = __int_as_float(0xFF800000); // -inf
    s[i] = 0.0f;
  }
}

// ---------- K1: e[E,H] = feats @ att^T via V_WMMA_F32_16X16X4_F32 ----------
// One wave32 per 16-edge tile. A = 16x4 f32 feature slice, fp32 accumulate
// across 16 K-steps (K=64 total).
// A layout (ISA 7.12.2): lanes 0-15 hold K=0(v0),K=1(v1); lanes 16-31 K=2,K=3.
// B (4x16, K x N): cols n = heads. Columns of D are independent, and we only
// read n<4, so lanes n>=4 load a duplicate valid att row (n&3) -> branch-free,
// no exec-mask juggling in the hot loop.
// C/D layout: VGPR v -> row v (lanes 0-15) / v+8 (lanes 16-31), col = lane%16.
__global__ __launch_bounds__(256)
void e_wmma_kernel(const float* __restrict__ feats, const float* __restrict__ att,
                   const int* __restrict__ dst, float* __restrict__ e_out,
                   float* __restrict__ m, long tiles) {
  long gtid = (long)blockIdx.x * blockDim.x + threadIdx.x;
  long wave = gtid >> 5;
  if (wave >= tiles) return;                 // wave-uniform: EXEC all-1 inside
  int lane = threadIdx.x & 31;
  int half = lane >> 4;                      // 0: K%4 in {0,1}; 1: K%4 in {2,3}
  int l16  = lane & 15;

  long edgeBase = wave * 16;
  const float* arow = feats + (edgeBase + l16) * F_DIM + half * 2;
  const float* brow = att + (l16 & (H_DIM - 1)) * F_DIM + half * 2;

  // Preload all 16 B K-slices (att is 1KB, loop-invariant): pure loads, no
  // branches, stays resident in 32 VGPRs for the whole tile.
  v2f b[16];
#pragma unroll
  for (int k = 0; k < 16; ++k) {
    b[k].x = brow[4 * k + 0];
    b[k].y = brow[4 * k + 1];
  }

  v8f c = {0.f, 0.f, 0.f, 0.f, 0.f, 0.f, 0.f, 0.f};
#pragma unroll
  for (int k = 0; k < 16; ++k) {
    // feats is streamed once -> non-temporal so 409MB doesn't thrash L2
    v2f a = __builtin_nontemporal_load((const v2f*)arow + 2 * k);
    c = __builtin_amdgcn_wmma_f32_16x16x4_f32(false, a, false, b[k],
                                              (short)0, c, false, false);
  }

  if (l16 < H_DIM) {                         // only head columns are valid
    int h = l16;
#pragma unroll
    for (int v = 0; v < 8; ++v) {
      long eidx = edgeBase + v + half * 8;   // row of D this VGPR holds
      float val = c[v];
      e_out[eidx * H_DIM + h] = val;
      int d = dst[eidx];
      atomic_max_f32(&m[(long)d * H_DIM + h], val);
    }
  }
}

// ---------- K1b: scalar tail for E % 16 != 0 (not hit for E=1.6M) ----------
__global__ __launch_bounds__(256)
void e_tail_kernel(const float* __restrict__ feats, const float* __restrict__ att,
                   const int* __restrict__ dst, float* __restrict__ e_out,
                   float* __restrict__ m, long start, long E) {
  long t = (long)blockIdx.x * blockDim.x + threadIdx.x;
  long eidx = start + (t >> 2);
  int h = (int)(t & 3);
  if (eidx >= E) return;
  const float* fr = feats + eidx * F_DIM;
  const float* ar = att + h * F_DIM;
  float acc = 0.0f;
#pragma unroll 8
  for (int k = 0; k < F_DIM; ++k) acc += fr[k] * ar[k];
  e_out[eidx * H_DIM + h] = acc;
  atomic_max_f32(&m[(long)dst[eidx] * H_DIM + h], acc);
}

// ---------- K2: ex = exp(e - m[dst]); s[dst] += ex ----------
__global__ __launch_bounds__(256)
void exp_sum_kernel(const int* __restrict__ dst, const float* __restrict__ m,
                    float* __restrict__ exb, float* __restrict__ s, long EH) {
  long i = (long)blockIdx.x * blockDim.x + threadIdx.x;
  if (i >= EH) return;
  long e = i >> 2;
  int h = (int)(i & 3);
  int d = dst[e];
  float v = __expf(exb[i] - m[(long)d * H_DIM + h]);
  exb[i] = v;
  atomic_add_f32(&s[(long)d * H_DIM + h], v);
}

// ---------- K3: w[e] = mean_h ex[e,h] / s[dst,h] ----------
__global__ __launch_bounds__(256)
void weight_kernel(const int* __restrict__ dst, const float* __restrict__ exb,
                   const float* __restrict__ s, float* __restrict__ wb, long E) {
  long e = (long)blockIdx.x * blockDim.x + threadIdx.x;
  if (e >= E) return;
  int d = dst[e];
  float w = 0.0f;
#pragma unroll
  for (int h = 0; h < H_DIM; ++h)
    w += exb[e * H_DIM + h] / s[(long)d * H_DIM + h];
  wb[e] = w * (1.0f / H_DIM);
}

// ---------- K4: out[dst,f] += w[e] * feats[e,f] ----------
__global__ __launch_bounds__(256)
void scatter_kernel(const float* __restrict__ feats, const int* __restrict__ dst,
                    const float* __restrict__ wb, float* __restrict__ out, long EF) {
  long i = (long)blockIdx.x * blockDim.x + threadIdx.x;
  if (i >= EF) return;
  long e = i >> 6;                // F_DIM == 64
  int f = (int)(i & 63);
  float w = wb[e];                // broadcast within 64-thread edge group
  int d = dst[e];
  float v = __builtin_nontemporal_load(&feats[e * F_DIM + f]); // stream once
  atomic_add_f32(&out[(long)d * F_DIM + f], w * v);            // L2-resident
}

extern "C" void kernel_launch(void* const* d_in, const int* in_sizes, int n_in,
                              void* d_out, int out_size, void* d_ws, size_t ws_size,
                              hipStream_t stream) {
  const float* feats = (const float*)d_in[0];   // [E, 64] f32
  const float* att   = (const float*)d_in[1];   // [1, 4, 64] f32
  const int*   dst   = (const int*)d_in[2];     // [E] i32
  // d_in[3] is n_nodes on device; derive N from out_size instead.

  const long E = in_sizes[2];
  const int  N = out_size / F_DIM;
  float* out = (float*)d_out;

  // workspace layout (floats): m[N*H] | s[N*H] | ex[E*H] | w[E]  (~35 MB)
  float* ws  = (float*)d_ws;
  float* m   = ws;
  float* s   = m + (size_t)N * H_DIM;
  float* exb = s + (size_t)N * H_DIM;
  float* wb  = exb + (size_t)E * H_DIM;

  const int NF = N * F_DIM, NH = N * H_DIM;
  init_kernel<<<(NF + 255) / 256, 256, 0, stream>>>(out, m, s, NF, NH);

  const long tiles = E / 16;
  if (tiles > 0) {
    long thr = tiles * 32;                     // one wave32 per 16-edge tile
    e_wmma_kernel<<<(int)((thr + 255) / 256), 256, 0, stream>>>(
        feats, att, dst, exb, m, tiles);
  }
  const long rem = E - tiles * 16;
  if (rem > 0) {
    long t = rem * H_DIM;
    e_tail_kernel<<<(int)((t + 255) / 256), 256, 0, stream>>>(
        feats, att, dst, exb, m, tiles * 16, E);
  }

  const long EH = E * H_DIM;
  exp_sum_kernel<<<(int)((EH + 255) / 256), 256, 0, stream>>>(dst, m, exb, s, EH);

  weight_kernel<<<(int)((E + 255) / 256), 256, 0, stream>>>(dst, exb, s, wb, E);

  const long EF = E * F_DIM;
  scatter_kernel<<<(int)((EF + 255) / 256), 256, 0, stream>>>(feats, dst, wb, out, EF);
}